// weighted_loss_89034672046818
// MI455X (gfx1250) — compile-verified
//
#include <hip/hip_runtime.h>

typedef float f32x4 __attribute__((ext_vector_type(4)));
typedef int   i32x4 __attribute__((ext_vector_type(4)));
typedef float v2f   __attribute__((ext_vector_type(2)));
typedef float v8f   __attribute__((ext_vector_type(8)));

// log2(1.1)
#define LOG2_1P1 0.13750352374993502f

#define NBLOCKS 2048   // stage-1 grid; partials = NBLOCKS floats in d_ws (8 KB)

// One channel's contribution: |p-t| * 1.1^|ais_p-ais_t| * (1 + piecewise(t,p))
__device__ __forceinline__ float chan_term(float p, float t,
                                           float T0, float T1, float T2, float T3, float T4,
                                           float a, float inv_a, float b, float inv_cb,
                                           float c, float d, float kk)
{
    int cp = (p >= T0) + (p >= T1) + (p >= T2) + (p >= T3) + (p >= T4);
    int ct = (t >= T0) + (t >= T1) + (t >= T2) + (t >= T3) + (t >= T4);
    float df   = fabsf((float)(cp - ct));
    float wcls = exp2f(LOG2_1P1 * df);            // 1.1^|dais|, one v_exp_f32

    float e = __expf(-kk * (t - d));              // only meaningful for t > d
    float w = (t <  0.0f) ? 0.0f
            : (t <  a)    ? t * inv_a
            : (t <= b)    ? 1.0f
            : (t <  c)    ? 1.0f - (t - b) * inv_cb
            : (t <= d)    ? 0.0f
            :               (e - 1.0f);
    w += (p < 0.0f) ? 1.0f : 0.0f;
    return fabsf(p - t) * wcls * (1.0f + w);
}

__device__ __forceinline__ float sample_term(float ph, float pc, float pn,
                                             float th, float tc, float tn, int ot)
{
    // head: a=80 b=1500 c=1750 d=2000, k=ln2/2000
    float r = chan_term(ph, th, 150.0f, 500.0f, 1000.0f, 1800.0f, 2600.0f,
                        80.0f, 1.0f/80.0f, 1500.0f, 1.0f/250.0f,
                        1750.0f, 2000.0f, 0.6931471805599453f/2000.0f);
    // chest: thresholds scaled by ot-dependent factor; a=10 b=75 c=85 d=100, k=ln2/100
    float s = (ot == 0) ? 0.8f : (ot == 1) ? 0.9f : (ot == 2) ? 1.0f
            : (ot == 3) ? 1.1f : 1.2f;
    r += chan_term(pc, tc, 22.0f*s, 35.0f*s, 45.0f*s, 55.0f*s, 65.0f*s,
                   10.0f, 1.0f/10.0f, 75.0f, 1.0f/10.0f,
                   85.0f, 100.0f, 0.6931471805599453f/100.0f);
    // neck: a=0.15 b=1.5 c=1.7 d=1.9, k=ln2/1.9
    r += chan_term(pn, tn, 0.2f, 0.5f, 1.0f, 1.5f, 2.0f,
                   0.15f, 1.0f/0.15f, 1.5f, 1.0f/0.19999999999999996f,
                   1.7f, 1.9f, 0.6931471805599453f/1.9f);
    return r;
}

// Stage 1: streaming grid-stride over groups of 4 samples (3x float4 per array),
// non-temporal b128 loads, per-block deterministic LDS tree reduction.
__global__ __launch_bounds__(256)
void loss_partial_kernel(const float* __restrict__ pred,
                         const float* __restrict__ tru,
                         const int*   __restrict__ ot,
                         float* __restrict__ partial,
                         int numGroups, int nSamples)
{
    const int tid    = blockIdx.x * blockDim.x + threadIdx.x;
    const int stride = gridDim.x * blockDim.x;

    const f32x4* __restrict__ p4 = (const f32x4*)pred;
    const f32x4* __restrict__ t4 = (const f32x4*)tru;
    const i32x4* __restrict__ o4 = (const i32x4*)ot;

    float acc = 0.0f;
    for (int g = tid; g < numGroups; g += stride) {
        size_t b3 = 3 * (size_t)g;
        f32x4 P0 = __builtin_nontemporal_load(&p4[b3 + 0]);
        f32x4 P1 = __builtin_nontemporal_load(&p4[b3 + 1]);
        f32x4 P2 = __builtin_nontemporal_load(&p4[b3 + 2]);
        f32x4 T0 = __builtin_nontemporal_load(&t4[b3 + 0]);
        f32x4 T1 = __builtin_nontemporal_load(&t4[b3 + 1]);
        f32x4 T2 = __builtin_nontemporal_load(&t4[b3 + 2]);
        i32x4 O  = __builtin_nontemporal_load(&o4[g]);

        acc += sample_term(P0[0], P0[1], P0[2], T0[0], T0[1], T0[2], O[0]);
        acc += sample_term(P0[3], P1[0], P1[1], T0[3], T1[0], T1[1], O[1]);
        acc += sample_term(P1[2], P1[3], P2[0], T1[2], T1[3], T2[0], O[2]);
        acc += sample_term(P2[1], P2[2], P2[3], T2[1], T2[2], T2[3], O[3]);
    }

    // Tail samples (B % 4 != 0), handled by global thread 0 (deterministic)
    if (tid == 0) {
        for (int i = numGroups * 4; i < nSamples; ++i) {
            size_t b = 3 * (size_t)i;
            acc += sample_term(pred[b], pred[b+1], pred[b+2],
                               tru[b],  tru[b+1],  tru[b+2], ot[i]);
        }
    }

    __shared__ float sm[256];
    sm[threadIdx.x] = acc;
    __syncthreads();
    #pragma unroll
    for (int s = 128; s > 0; s >>= 1) {
        if (threadIdx.x < s) sm[threadIdx.x] += sm[threadIdx.x + s];
        __syncthreads();
    }
    if (threadIdx.x == 0) partial[blockIdx.x] = sm[0];
}

// Stage 2: one wave32. Lane-sums NP partials (compile-time count => unrolled,
// batched loads), then reduces across lanes with V_WMMA_F32_16X16X4_F32:
// D = A x ones => every D element is a row-sum of A, so sum(all D)/16 ==
// sum(all A) exactly (/16 is a power of two — no precision loss).
template <int NP>
__global__ __launch_bounds__(32)
void loss_final_kernel(const float* __restrict__ partial,
                       float* __restrict__ out,
                       float invB)
{
    const int lane = threadIdx.x;
    float s = 0.0f;
    #pragma unroll
    for (int i = 0; i < NP / 32; ++i) s += partial[lane + 32 * i];

    v2f A;  A[0] = s;     A[1] = 0.0f;   // lane partial in A's first VGPR
    v2f Bm; Bm[0] = 1.0f; Bm[1] = 1.0f;  // B = all ones (layout-invariant)
    v8f C = {};
    C = __builtin_amdgcn_wmma_f32_16x16x4_f32(
            /*neg_a=*/false, A, /*neg_b=*/false, Bm,
            /*c_mod=*/(short)0, C, /*reuse_a=*/false, /*reuse_b=*/false);

    float r = C[0] + C[1] + C[2] + C[3] + C[4] + C[5] + C[6] + C[7];

    __shared__ float sm[32];
    sm[lane] = r;
    __syncthreads();
    if (lane == 0) {
        float tot = 0.0f;
        #pragma unroll
        for (int i = 0; i < 32; ++i) tot += sm[i];
        out[0] = tot * 0.0625f * invB;   // /16 (exact) then /B
    }
}

extern "C" void kernel_launch(void* const* d_in, const int* in_sizes, int n_in,
                              void* d_out, int out_size, void* d_ws, size_t ws_size,
                              hipStream_t stream)
{
    const float* pred = (const float*)d_in[0];
    const float* tru  = (const float*)d_in[1];
    const int*   ot   = (const int*)d_in[2];
    float* out        = (float*)d_out;
    float* partial    = (float*)d_ws;    // NBLOCKS floats = 8 KB of scratch

    const int B         = in_sizes[0] / 3;
    const int numGroups = B / 4;

    loss_partial_kernel<<<NBLOCKS, 256, 0, stream>>>(pred, tru, ot, partial,
                                                     numGroups, B);
    loss_final_kernel<NBLOCKS><<<1, 32, 0, stream>>>(partial, out, 1.0f / (float)B);
}